// GridLayer_11141145166394
// MI455X (gfx1250) — compile-verified
//
#include <hip/hip_runtime.h>
#include <cstdint>

// Problem constants from the reference
#define B_ 4
#define N_ 65536
#define E_ 128         // floats per row (512 bytes)
#define NH_ 7
#define GLOBAL_LEVEL_ 0

// gfx12+ CPol immediate: TH in bits [2:0], scope in bits [4:3].
// Store TH: 0 = RT (regular temporal), 1 = NT (non-temporal/streaming).
#define CPOL_RT 0
#define CPOL_NT 1

typedef float v4f __attribute__((ext_vector_type(4)));
typedef int   v4i __attribute__((ext_vector_type(4)));

#if defined(__AMDGCN__) &&                                              \
    __has_builtin(__builtin_amdgcn_global_load_async_to_lds_b128) &&    \
    __has_builtin(__builtin_amdgcn_global_store_async_from_lds_b128)
#define HAVE_ASYNC 1
#else
#define HAVE_ASYNC 0
#endif

// Builtin parameter types (from hipcc diagnostic): pointers to 4xint vectors.
typedef __attribute__((address_space(1))) v4i* gptr_v4i;
typedef __attribute__((address_space(3))) v4i* lptr_v4i;

__device__ __forceinline__ gptr_v4i to_g(const void* p) {
  // generic global address == AS1 address (64-bit)
  return (gptr_v4i)(uintptr_t)p;
}
__device__ __forceinline__ lptr_v4i to_l(const void* p) {
  // low 32 bits of a generic LDS address are the LDS byte offset
  return (lptr_v4i)(uint32_t)(uintptr_t)p;
}

// One wave32 per (b, n) cell: copies NH_ neighbor rows of 512B each,
// 16B per lane, via the CDNA5 async global<->LDS DMA path.
__global__ __launch_bounds__(256) void gather_nh_kernel(
    const float* __restrict__ x,                       // [B,N,1,E]
    const int* __restrict__ adjc,                      // [N,NH]
    const unsigned char* __restrict__ adjc_mask_inv,   // [N,NH] bool
    const int* __restrict__ local_indices,             // [B,N]
    const int* __restrict__ batch_sample_indices,      // [B]
    const int* __restrict__ sampled_level_p,           // [1]
    float* __restrict__ out_x,                         // [B,N,NH,1,E]
    float* __restrict__ out_mask)                      // [B,N,NH,1] as float
{
  const int tid       = threadIdx.x;
  const int lane      = tid & 31;
  const int waveInBlk = tid >> 5;
  const int p = blockIdx.x * 8 + waveInBlk;  // cell id; grid sized so p < B_*N_
  const int b = p >> 16;                     // N_ == 65536
  // n = p & (N_-1)  (implicit in p)

  const int li   = local_indices[p];         // uniform across the wave
  const int slvl = sampled_level_p[0];
  const long long off =
      (long long)batch_sample_indices[b] << (2 * (slvl - GLOBAL_LEVEL_));

  int idx[NH_];
#pragma unroll
  for (int h = 0; h < NH_; ++h)
    idx[h] = adjc[li * NH_ + h] - (int)off;

  // Tiny mask output: lanes 0..6 each write one float (bool -> 0.0/1.0)
  if (lane < NH_) {
    out_mask[(size_t)p * NH_ + lane] =
        adjc_mask_inv[li * NH_ + lane] ? 1.0f : 0.0f;
  }

  const size_t xbase = (size_t)b * N_ * E_;
  float* dst0 = out_x + (size_t)p * NH_ * E_;

#if HAVE_ASYNC
  // Per-wave staging buffer: NH_ rows * 512B = 3.5KB; 8 waves -> 28KB/block
  // (of the 320KB WGP pool -> no occupancy limit).
  __shared__ float lbuf[8 * NH_ * E_];
  float* myb = &lbuf[waveInBlk * (NH_ * E_)];

  // Issue all 7 row loads (memory -> LDS), 16B per lane, fully async.
  // TH=RT: x rows have ~7x reuse and fit in the 192MB L2 -> let them cache.
#pragma unroll
  for (int h = 0; h < NH_; ++h) {
    const float* src = x + xbase + (size_t)idx[h] * E_ + lane * 4;
    __builtin_amdgcn_global_load_async_to_lds_b128(
        to_g(src), to_l(&myb[h * E_ + lane * 4]), 0, CPOL_RT);
  }

  // Single bulk wait: all loads landed in LDS.
#if __has_builtin(__builtin_amdgcn_s_wait_asynccnt)
  __builtin_amdgcn_s_wait_asynccnt(0);
#else
  asm volatile("s_wait_asynccnt 0" ::: "memory");
#endif

  // Fire-and-forget streaming stores (LDS -> memory); TH=NT so the 940MB
  // write-once output stream does not evict x from L2. Drained by the
  // implicit wait-idle at S_ENDPGM.
#pragma unroll
  for (int h = 0; h < NH_; ++h) {
    __builtin_amdgcn_global_store_async_from_lds_b128(
        to_g(dst0 + h * E_ + lane * 4), to_l(&myb[h * E_ + lane * 4]),
        0, CPOL_NT);
  }
#else
  // Fallback: 128-bit loads + non-temporal 128-bit stores.
#pragma unroll
  for (int h = 0; h < NH_; ++h) {
    const v4f v = *(const v4f*)(x + xbase + (size_t)idx[h] * E_ + lane * 4);
    __builtin_nontemporal_store(v, (v4f*)(dst0 + h * E_ + lane * 4));
  }
#endif
}

extern "C" void kernel_launch(void* const* d_in, const int* in_sizes, int n_in,
                              void* d_out, int out_size, void* d_ws,
                              size_t ws_size, hipStream_t stream) {
  const float* x          = (const float*)d_in[0];
  const int* adjc         = (const int*)d_in[1];
  const unsigned char* mi = (const unsigned char*)d_in[2];  // bool buffer
  const int* lidx         = (const int*)d_in[3];
  const int* bsi          = (const int*)d_in[4];
  const int* slvl         = (const int*)d_in[5];

  float* out_x    = (float*)d_out;
  float* out_mask = out_x + (size_t)B_ * N_ * NH_ * E_;  // 234,881,024 floats

  // 8 waves per block, one (b,n) cell per wave: exactly B_*N_ waves.
  const int blocks = (B_ * N_) / 8;  // 32768
  gather_nh_kernel<<<blocks, 256, 0, stream>>>(x, adjc, mi, lidx, bsi, slvl,
                                               out_x, out_mask);
}